// ParallelHyenaConv_81277961110220
// MI455X (gfx1250) — compile-verified
//
#include <hip/hip_runtime.h>

typedef float v2f __attribute__((ext_vector_type(2)));
typedef float v8f __attribute__((ext_vector_type(8)));

#define L_LEN 2048
#define N_HEAD 128
#define D_MODEL_SZ 1024
#define HEAD_DIM 8
#define CHUNK 512
#define ROWS_PER_BLK 128
#define NWAVE 8
#define KPAD 16

// Hyena "attention form":  Out(j,l) = sum_{t<=l} (kappa(l-t) + D*delta(l,t)) * S(l,t) * v_j(t)
// with S(l,t) = sum_i q_i(l) k_i(t).  All GEMMs via V_WMMA_F32_16X16X4_F32.
//  - kappa is staged zero-padded (negative lags read 0) with D folded into kappa(0)
//  - V is staged with 8 zero rows so the j-dimension needs no masking
//  - next chunk's K/V are prefetched (global_prefetch) while computing the current chunk
__global__ __launch_bounds__(256)
void hyena_attn_kernel(const float* __restrict__ kg,
                       const float* __restrict__ vg,
                       const float* __restrict__ qg,
                       const float* __restrict__ kerg,
                       const float* __restrict__ Dg,
                       float* __restrict__ outg)
{
    __shared__ __attribute__((aligned(16))) float sK[HEAD_DIM][CHUNK];        // 16 KB
    __shared__ __attribute__((aligned(16))) float sVp[16][CHUNK];             // 32 KB (rows 8..15 zero)
    __shared__ __attribute__((aligned(16))) float sQ[HEAD_DIM][ROWS_PER_BLK]; // 4 KB
    __shared__ __attribute__((aligned(16))) float sKerP[KPAD + L_LEN];        // 8.06 KB
    __shared__ __attribute__((aligned(16))) float sW[NWAVE][16 * 17];         // 8.5 KB

    const int tid  = threadIdx.x;
    const int lane = tid & 31;
    const int wid  = tid >> 5;

    const int rblk = blockIdx.x & 15;          // L / ROWS_PER_BLK == 16 row-blocks
    const int bh   = blockIdx.x >> 4;
    const int b    = bh / N_HEAD;
    const int h    = bh % N_HEAD;

    const size_t headBase = ((size_t)b * D_MODEL_SZ + (size_t)h * HEAD_DIM) * L_LEN;

    const float Dh = Dg[h];

    // ---- stage Q tile (8 x 128), padded kappa, zero V-pad rows ----
    {
        int i  = tid >> 5;                      // row 0..7
        int c4 = tid & 31;                      // 32 float4 per 128-float row
        const float4* src = (const float4*)(qg + headBase + (size_t)i * L_LEN + rblk * ROWS_PER_BLK);
        ((float4*)&sQ[i][0])[c4] = src[c4];
    }
    {
        const float4* src = (const float4*)(kerg + (size_t)h * L_LEN);
        for (int u = tid; u < L_LEN / 4; u += 256) {
            float4 f = src[u];
            if (u == 0) f.x += Dh;              // fold D-skip into kappa(0)
            ((float4*)(sKerP + KPAD))[u] = f;
        }
        if (tid < KPAD) sKerP[tid] = 0.0f;      // negative lags -> 0
    }
    {
        float4 z = make_float4(0.f, 0.f, 0.f, 0.f);
        for (int u = tid; u < 8 * CHUNK / 4; u += 256)
            ((float4*)&sVp[8][0])[u] = z;       // zero rows 8..15 once
    }
    __syncthreads();

    const int rowBase  = rblk * ROWS_PER_BLK + wid * 16;  // this wave's 16 l-rows
    const int rowBaseU = __builtin_amdgcn_readfirstlane(rowBase); // provably uniform
    const int n  = lane & 15;                  // N index / A-matrix row
    const int k0 = (lane < 16) ? 0 : 2;        // K lane-half pattern (A and B)
    const int mB = (lane < 16) ? 0 : 8;        // C/D M base per lane-half

    // Q tile in A-layout (M=l, K=i), two K=4 chunks — loop invariant
    v2f qa0, qa1;
    {
        int lcol = wid * 16 + n;
        qa0.x = sQ[k0 + 0][lcol];
        qa0.y = sQ[k0 + 1][lcol];
        qa1.x = sQ[4 + k0 + 0][lcol];
        qa1.y = sQ[4 + k0 + 1][lcol];
    }

    v8f acc = {};                               // Out tile (l x j), f32
    float* sWw = &sW[wid][0];
    const int arow = n * 17;

    const int nChunk = (rblk >> 2) + 1;         // chunk c needed iff 4c <= rblk (causal)
    for (int c = 0; c < nChunk; ++c) {
        const int tStart = c * CHUNK;

        // ---- stage K,V chunk (8 x 512 each; V rows 8..15 stay zero) ----
        for (int u = tid; u < HEAD_DIM * CHUNK / 4; u += 256) {
            int i  = u >> 7;                    // 128 float4 per row
            int c4 = u & 127;
            ((float4*)&sK[i][0])[c4]  = ((const float4*)(kg + headBase + (size_t)i * L_LEN + tStart))[c4];
            ((float4*)&sVp[i][0])[c4] = ((const float4*)(vg + headBase + (size_t)i * L_LEN + tStart))[c4];
        }

        // ---- prefetch next chunk's K/V into cache while we compute this one ----
        if (c + 1 < nChunk) {
            const int tNext = tStart + CHUNK;
            const float* basePtr = (tid < 128) ? kg : vg;
            int r  = (tid & 127) >> 4;          // row 0..7
            int ln = (tid & 15) * 32;           // 16 x 128B segments per 2KB row
            __builtin_prefetch(basePtr + headBase + (size_t)r * L_LEN + tNext + ln, 0, 3);
        }
        __syncthreads();

        // scalar causal trip count: tiles with colBase <= rowBase
        int ttEnd = ((rowBaseU - tStart) >> 4) + 1;
        if (ttEnd > CHUNK / 16) ttEnd = CHUNK / 16;

        for (int tt = 0; tt < ttEnd; ++tt) {
            const int colLoc  = tt * 16;
            const int colBase = tStart + colLoc;

            // ---- S = Q^T K : 16x16, inner dim 8 -> 2 WMMAs ----
            v2f bk0, bk1;
            bk0.x = sK[k0 + 0][colLoc + n];
            bk0.y = sK[k0 + 1][colLoc + n];
            bk1.x = sK[4 + k0 + 0][colLoc + n];
            bk1.y = sK[4 + k0 + 1][colLoc + n];

            v8f s = {};
            s = __builtin_amdgcn_wmma_f32_16x16x4_f32(false, qa0, false, bk0, (short)0, s, false, false);
            s = __builtin_amdgcn_wmma_f32_16x16x4_f32(false, qa1, false, bk1, (short)0, s, false, false);

            // ---- Toeplitz/causal modulation (branchless: padded kappa) ----
            const int kerBase = KPAD + rowBase + mB - (colBase + n); // >= 1
            #pragma unroll
            for (int r = 0; r < 8; ++r) {
                float wc = sKerP[kerBase + r];
                sWw[(mB + r) * 17 + n] = s[r] * wc;
            }

            // ---- Out += W * V : inner dim t=16 -> 4 WMMAs (V rows >=8 are zero) ----
            #pragma unroll
            for (int kc = 0; kc < 4; ++kc) {
                v2f aw, bv;
                aw.x = sWw[arow + 4 * kc + k0 + 0];
                aw.y = sWw[arow + 4 * kc + k0 + 1];
                int tl = colLoc + 4 * kc + k0;
                bv.x = sVp[n][tl + 0];
                bv.y = sVp[n][tl + 1];
                acc = __builtin_amdgcn_wmma_f32_16x16x4_f32(false, aw, false, bv, (short)0, acc, false, false);
            }
        }
        __syncthreads();   // protect sK/sVp before next chunk's staging
    }

    // ---- store: acc lane element r -> out[b, h*8+n, rowBase+mB+r], r=0..7 contiguous in l
    if (n < 8) {
        size_t o = headBase + (size_t)n * L_LEN + (size_t)(rowBase + mB);
        float4 lo = make_float4(acc[0], acc[1], acc[2], acc[3]);
        float4 hi = make_float4(acc[4], acc[5], acc[6], acc[7]);
        *(float4*)(outg + o)     = lo;
        *(float4*)(outg + o + 4) = hi;
    }
}

extern "C" void kernel_launch(void* const* d_in, const int* in_sizes, int n_in,
                              void* d_out, int out_size, void* d_ws, size_t ws_size,
                              hipStream_t stream) {
    const float* k   = (const float*)d_in[0];
    const float* v   = (const float*)d_in[1];
    const float* q   = (const float*)d_in[2];
    const float* ker = (const float*)d_in[3];
    const float* D   = (const float*)d_in[4];
    float* out = (float*)d_out;

    const int B = in_sizes[0] / (D_MODEL_SZ * L_LEN);   // 2
    dim3 grid((unsigned)(B * N_HEAD * (L_LEN / ROWS_PER_BLK)));  // 4096 blocks
    hipLaunchKernelGGL(hyena_attn_kernel, grid, dim3(256), 0, stream, k, v, q, ker, D, out);
}